// LGCN_Net_87634512707727
// MI455X (gfx1250) — compile-verified
//
#include <hip/hip_runtime.h>
#include <hip/hip_bf16.h>
#include <math.h>

typedef float v2f __attribute__((ext_vector_type(2)));
typedef float v8f __attribute__((ext_vector_type(8)));

#define F_INC 64
#define KCUT 16
#define CCH 20
#define LL 4
#define H1C 20
#define H2C 2

__device__ __forceinline__ int rfl_i(int v) {
    return __builtin_amdgcn_readfirstlane(v);
}
__device__ __forceinline__ float rfl_f(float v) {
    return __int_as_float(__builtin_amdgcn_readfirstlane(__float_as_int(v)));
}

// ---------------- Filter net: per original edge (mirrored edges share attr) ----
__global__ void filter_net_kernel(const float* __restrict__ attr,
                                  const float* __restrict__ cut,
                                  const float* __restrict__ cw,
                                  const float* __restrict__ cb,
                                  const float* __restrict__ f1w,
                                  const float* __restrict__ f1b,
                                  const float* __restrict__ f2w,
                                  const float* __restrict__ f2b,
                                  float* __restrict__ wout,
                                  int E)
{
    int e = blockIdx.x * blockDim.x + threadIdx.x;
    if (e >= E) return;
    float a = attr[e];
    // padded 2-channel signal: dd = attr - cutoff, ss = (dd>0)
    float dd[KCUT + 2], ss[KCUT + 2];
    dd[0] = 0.f; ss[0] = 0.f; dd[KCUT + 1] = 0.f; ss[KCUT + 1] = 0.f;
    #pragma unroll
    for (int k = 0; k < KCUT; k++) {
        float d = a - cut[k];
        dd[k + 1] = d;
        ss[k + 1] = d > 0.f ? 1.f : 0.f;
    }
    // conv(2->20,k=3,pad=1) + max over K + relu, fused into fc1 accumulation
    float h1a[2 * LL];
    #pragma unroll
    for (int o = 0; o < 2 * LL; o++) h1a[o] = f1b[o];
    for (int c = 0; c < CCH; c++) {
        float w0 = cw[c * 6 + 0], w1 = cw[c * 6 + 1], w2 = cw[c * 6 + 2];
        float u0 = cw[c * 6 + 3], u1 = cw[c * 6 + 4], u2 = cw[c * 6 + 5];
        float bb = cb[c];
        float mx = -1e30f;
        #pragma unroll
        for (int k = 0; k < KCUT; k++) {
            float acc = bb + w0 * dd[k] + w1 * dd[k + 1] + w2 * dd[k + 2]
                           + u0 * ss[k] + u1 * ss[k + 1] + u2 * ss[k + 2];
            mx = fmaxf(mx, acc);
        }
        mx = fmaxf(mx, 0.f);   // ReLU(AdaptiveMaxPool)
        #pragma unroll
        for (int o = 0; o < 2 * LL; o++) h1a[o] += f1w[o * CCH + c] * mx;
    }
    #pragma unroll
    for (int o = 0; o < 2 * LL; o++) h1a[o] = fmaxf(h1a[o], 0.f);
    float4 wo;
    float* wop = (float*)&wo;
    #pragma unroll
    for (int l = 0; l < LL; l++) {
        float acc = f2b[l];
        #pragma unroll
        for (int o = 0; o < 2 * LL; o++) acc += f2w[l * 2 * LL + o] * h1a[o];
        wop[l] = fmaxf(acc, 0.f);
    }
    ((float4*)wout)[e] = wo;
}

// ---------------- scratch zeroing (float4-wide) ------------------------------
__global__ void zero_kernel(float4* __restrict__ p, long long n4)
{
    long long i = (long long)blockIdx.x * blockDim.x + threadIdx.x;
    if (i < n4) p[i] = make_float4(0.f, 0.f, 0.f, 0.f);
}

// ---------------- layer-1 scatter: ONE WAVE per (bidirectional) edge ----------
// e2 is wave-uniform -> endpoints + weights become scalar loads; each lane
// covers 2 of the 64 features (coalesced float2 read of x[src], 8 atomics).
__global__ void scatter1_kernel(const int* __restrict__ ei,
                                const float* __restrict__ x,
                                const float* __restrict__ w,
                                float* __restrict__ agg,
                                int E, int N)
{
    int wave = blockIdx.x * (blockDim.x >> 5) + (threadIdx.x >> 5);
    int lane = threadIdx.x & 31;
    int e2 = rfl_i(wave);                 // wave-uniform edge id in SGPR
    if (e2 >= 2 * E) return;
    int e = e2 < E ? e2 : e2 - E;
    int si, di;
    if (e2 < E) { si = ei[e];     di = ei[E + e]; }
    else        { si = ei[E + e]; di = ei[e];     }
    si = rfl_i(si);
    di = rfl_i(di);
    float4 wv = ((const float4*)w)[e];    // mirrored edge reuses same filter out
    float w0 = rfl_f(wv.x), w1 = rfl_f(wv.y), w2 = rfl_f(wv.z), w3 = rfl_f(wv.w);

    int f = lane << 1;
    float2 xv = *(const float2*)(x + (size_t)si * F_INC + f);
    float* b0 = agg + (size_t)di * F_INC + f;
    size_t lstride = (size_t)N * F_INC;
    unsafeAtomicAdd(b0 + 0,               w0 * xv.x);
    unsafeAtomicAdd(b0 + 1,               w0 * xv.y);
    unsafeAtomicAdd(b0 + lstride + 0,     w1 * xv.x);
    unsafeAtomicAdd(b0 + lstride + 1,     w1 * xv.y);
    unsafeAtomicAdd(b0 + 2 * lstride + 0, w2 * xv.x);
    unsafeAtomicAdd(b0 + 2 * lstride + 1, w2 * xv.y);
    unsafeAtomicAdd(b0 + 3 * lstride + 0, w3 * xv.x);
    unsafeAtomicAdd(b0 + 3 * lstride + 1, w3 * xv.y);
}

// ---------------- layer-1 fused GEMM via fp32 WMMA ---------------------------
// h[n,:] = relu( [x | agg0..3][n,:] (K=320) @ [root1; theta1[0..3]] + bias1 )
// One wave computes one 16x16 output tile; 80 V_WMMA_F32_16X16X4_F32 per tile.
__global__ void __launch_bounds__(32)
gemm1_wmma_kernel(const float* __restrict__ x,
                  const float* __restrict__ agg,
                  const float* __restrict__ root1,
                  const float* __restrict__ theta1,
                  const float* __restrict__ bias1,
                  float* __restrict__ h,
                  int N)
{
    int tile = blockIdx.x;
    int n0 = (tile & 1) * 16;
    int m0 = (tile >> 1) * 16;
    int lane = threadIdx.x;
    int half = lane >> 4;      // K sub-pair select (ISA 16x4 A layout)
    int lm   = lane & 15;      // M row (A) / N col (B,D)
    int ncol = n0 + lm;
    int ccol = ncol < H1C ? ncol : 0;
    float bmask = ncol < H1C ? 1.f : 0.f;   // branchless N=20 padding to 32

    v8f acc = {0.f, 0.f, 0.f, 0.f, 0.f, 0.f, 0.f, 0.f};
    for (int seg = 0; seg < 5; seg++) {
        const float* Asrc = (seg == 0) ? x     : (agg    + (long long)(seg - 1) * N * F_INC);
        const float* Bsrc = (seg == 0) ? root1 : (theta1 + (seg - 1) * F_INC * H1C);
        const float* arow = Asrc + (long long)(m0 + lm) * F_INC;
        #pragma unroll
        for (int kb = 0; kb < F_INC; kb += 4) {
            int k = kb + 2 * half;
            v2f a; a.x = arow[k];                    a.y = arow[k + 1];
            v2f b; b.x = Bsrc[k * H1C + ccol] * bmask;
                   b.y = Bsrc[(k + 1) * H1C + ccol] * bmask;
            acc = __builtin_amdgcn_wmma_f32_16x16x4_f32(
                false, a, false, b, (short)0, acc, false, false);
        }
    }
    if (ncol < H1C) {
        float bv = bias1[ncol];
        #pragma unroll
        for (int i = 0; i < 8; i++) {
            int row = m0 + half * 8 + i;             // D layout: VGPR i -> M=i / M=8+i
            float v = acc[i] + bv;
            h[(long long)row * H1C + ncol] = v > 0.f ? v : 0.f;  // relu(h)
        }
    }
}

// ---------------- layer-2 scatter: ONE WAVE per edge (lanes 0..19 active) -----
__global__ void scatter2_kernel(const int* __restrict__ ei,
                                const float* __restrict__ h,
                                const float* __restrict__ w,
                                float* __restrict__ agg2,
                                int E, int N)
{
    int wave = blockIdx.x * (blockDim.x >> 5) + (threadIdx.x >> 5);
    int lane = threadIdx.x & 31;
    int e2 = rfl_i(wave);
    if (e2 >= 2 * E) return;
    int e = e2 < E ? e2 : e2 - E;
    int si, di;
    if (e2 < E) { si = ei[e];     di = ei[E + e]; }
    else        { si = ei[E + e]; di = ei[e];     }
    si = rfl_i(si);
    di = rfl_i(di);
    float4 wv = ((const float4*)w)[e];
    float w0 = rfl_f(wv.x), w1 = rfl_f(wv.y), w2 = rfl_f(wv.z), w3 = rfl_f(wv.w);
    if (lane >= H1C) return;

    float hv = h[(size_t)si * H1C + lane];
    float* b0 = agg2 + (size_t)di * H1C + lane;
    size_t lstride = (size_t)N * H1C;
    unsafeAtomicAdd(b0,               w0 * hv);
    unsafeAtomicAdd(b0 + lstride,     w1 * hv);
    unsafeAtomicAdd(b0 + 2 * lstride, w2 * hv);
    unsafeAtomicAdd(b0 + 3 * lstride, w3 * hv);
}

// ---------------- layer-2 output GEMM (K=100, Nout=2) + log_softmax ----------
__global__ void out2_kernel(const float* __restrict__ h,
                            const float* __restrict__ agg2,
                            const float* __restrict__ root2,
                            const float* __restrict__ theta2,
                            const float* __restrict__ bias2,
                            float* __restrict__ out,
                            int N)
{
    int n = blockIdx.x * blockDim.x + threadIdx.x;
    if (n >= N) return;
    float o0 = bias2[0], o1 = bias2[1];
    const float* hp = h + (long long)n * H1C;
    #pragma unroll
    for (int k = 0; k < H1C; k++) {
        float hv = hp[k];
        o0 += hv * root2[k * 2 + 0];
        o1 += hv * root2[k * 2 + 1];
    }
    for (int l = 0; l < LL; l++) {
        const float* ap = agg2 + ((long long)l * N + n) * H1C;
        const float* tp = theta2 + l * H1C * 2;
        #pragma unroll
        for (int k = 0; k < H1C; k++) {
            float av = ap[k];
            o0 += av * tp[k * 2 + 0];
            o1 += av * tp[k * 2 + 1];
        }
    }
    float m = fmaxf(o0, o1);
    float lse = m + logf(expf(o0 - m) + expf(o1 - m));
    out[(long long)n * 2 + 0] = o0 - lse;
    out[(long long)n * 2 + 1] = o1 - lse;
}

extern "C" void kernel_launch(void* const* d_in, const int* in_sizes, int n_in,
                              void* d_out, int out_size, void* d_ws, size_t ws_size,
                              hipStream_t stream)
{
    const float* x     = (const float*)d_in[0];
    const int*   ei    = (const int*)  d_in[1];
    const float* attr  = (const float*)d_in[2];
    const float* cut   = (const float*)d_in[3];
    const float* cw1   = (const float*)d_in[4];
    const float* cb1   = (const float*)d_in[5];
    const float* f1w1  = (const float*)d_in[6];
    const float* f1b1  = (const float*)d_in[7];
    const float* f2w1  = (const float*)d_in[8];
    const float* f2b1  = (const float*)d_in[9];
    const float* th1   = (const float*)d_in[10];
    const float* rt1   = (const float*)d_in[11];
    const float* bi1   = (const float*)d_in[12];
    const float* cw2   = (const float*)d_in[13];
    const float* cb2   = (const float*)d_in[14];
    const float* f1w2  = (const float*)d_in[15];
    const float* f1b2  = (const float*)d_in[16];
    const float* f2w2  = (const float*)d_in[17];
    const float* f2b2  = (const float*)d_in[18];
    const float* th2   = (const float*)d_in[19];
    const float* rt2   = (const float*)d_in[20];
    const float* bi2   = (const float*)d_in[21];

    const int N = in_sizes[0] / F_INC;   // 100000
    const int E = in_sizes[2];           // 1600000

    // workspace layout (bytes): [w: E*4 f32][agg: 4*N*64 f32, reused as 4*N*20][h: N*20 f32]
    char*  ws   = (char*)d_ws;
    float* wbuf = (float*)ws;
    float* agg  = (float*)(ws + (size_t)E * 4 * sizeof(float));
    float* hbuf = (float*)(ws + (size_t)E * 4 * sizeof(float)
                              + (size_t)4 * N * F_INC * sizeof(float));

    const int bs = 256;
    const int wavesPerBlock = bs / 32;

    // ---- layer 1 ----
    filter_net_kernel<<<(E + bs - 1) / bs, bs, 0, stream>>>(
        attr, cut, cw1, cb1, f1w1, f1b1, f2w1, f2b1, wbuf, E);

    long long n4a = (long long)4 * N * F_INC / 4;
    zero_kernel<<<(unsigned)((n4a + bs - 1) / bs), bs, 0, stream>>>((float4*)agg, n4a);

    long long waves1 = (long long)2 * E;                 // one wave per edge
    scatter1_kernel<<<(unsigned)((waves1 + wavesPerBlock - 1) / wavesPerBlock), bs, 0, stream>>>(
        ei, x, wbuf, agg, E, N);

    gemm1_wmma_kernel<<<(unsigned)((N / 16) * 2), 32, 0, stream>>>(
        x, agg, rt1, th1, bi1, hbuf, N);

    // ---- layer 2 ----
    filter_net_kernel<<<(E + bs - 1) / bs, bs, 0, stream>>>(
        attr, cut, cw2, cb2, f1w2, f1b2, f2w2, f2b2, wbuf, E);

    long long n4b = (long long)4 * N * H1C / 4;
    zero_kernel<<<(unsigned)((n4b + bs - 1) / bs), bs, 0, stream>>>((float4*)agg, n4b);

    long long waves2 = (long long)2 * E;                 // one wave per edge
    scatter2_kernel<<<(unsigned)((waves2 + wavesPerBlock - 1) / wavesPerBlock), bs, 0, stream>>>(
        ei, hbuf, wbuf, agg, E, N);

    out2_kernel<<<(N + bs - 1) / bs, bs, 0, stream>>>(
        hbuf, agg, rt2, th2, bi2, (float*)d_out, N);
}